// JPEGDecoder_89524298317930
// MI455X (gfx1250) — compile-verified
//
#include <hip/hip_runtime.h>

// ---------------------------------------------------------------------------
// JPEG inverse-DCT decoder for MI455X (gfx1250, wave32).
//
// Roofline: ~201 MB HBM traffic, 0.8 GFLOP -> pure bandwidth problem
// (~8.6 us floor at 23.3 TB/s). One streaming pass: coalesced b128 input
// staging into wave-private LDS, zigzag gather, the 8x8 DCT pair done as two
// 16x16x16 fp32 WMMA chains (v_wmma_f32_16x16x4_f32) on a block-diagonal
// DCT matrix (4 patches per 16x16 tile), fully contiguous 128B row stores.
//
// v2: 8 tiles per wave (amortizes the ~240-instruction per-wave setup so
// instruction issue stays well under the HBM roofline) + 1-deep prefetch
// pipeline: tile i+1's global loads are issued before tile i's LDS barrier,
// hiding HBM latency behind the WMMA/LDS work. All per-tile address deltas
// are compile-time immediates (fit in 24-bit signed ioffset).
// ---------------------------------------------------------------------------

typedef __attribute__((ext_vector_type(2))) float v2f;
typedef __attribute__((ext_vector_type(8))) float v8f;

// zigzag index table: patch(r,c) = coeff[ZZ[r*8+c]]  (matches _zigzag_idx)
__constant__ int ZZ[64] = {
     0,  1,  5,  6, 14, 15, 27, 28,
     2,  4,  7, 13, 16, 26, 29, 42,
     3,  8, 12, 17, 25, 30, 41, 43,
     9, 11, 18, 24, 31, 40, 44, 53,
    10, 19, 23, 32, 39, 45, 52, 54,
    20, 22, 33, 38, 46, 51, 55, 58,
    21, 34, 37, 47, 50, 56, 59, 61,
    35, 36, 48, 49, 57, 60, 62, 63};

// D[r][c] = sqrt(2/8) * cr * cos(pi*(c+0.5)*r/8), cr = 1/sqrt2 for r==0
__device__ __forceinline__ float dctv(int r, int c) {
    if (r == 0) return 0.35355339059327373f;
    // pi*(c+0.5)*r/8 = (pi/16)*(2c+1)*r
    return 0.5f * __cosf(0.19634954084936207f * (float)((2 * c + 1) * r));
}

// One wave handles 8 consecutive 16x16 tiles; each tile = 4 horizontally
// adjacent 8x8 patches (same b, ch, nr; patch-cols ncg*4 .. ncg*4+3).
// Tile blocks: X = [[P0,P1],[P2,P3]], q = (row/8)*2 + (col/8).
#define TILES_PER_WAVE 8
#define IN_TILE_STRIDE  (4 * 32 * 192)   // floats between 4-patch groups
#define OUT_TILE_STRIDE 32               // floats between 32-pixel strips

__global__ __launch_bounds__(256) void jpeg_idct_wmma(
        const float* __restrict__ comp,   // (N=4096, B=32, 192)
        float* __restrict__ out) {        // (32, 3, 512, 512)
    __shared__ float smem[8 * 512];       // per wave: 256 raw coeffs + 256 Y

    const int lane = threadIdx.x & 31;
    const int wv   = threadIdx.x >> 5;    // wave in block (wave32)
    const int half = lane >> 4;           // lane half (0/1)
    const int n16  = lane & 15;

    const int tile0 = (blockIdx.x * 8 + wv) * TILES_PER_WAVE; // 0..98296
    const int ncg0 = tile0 & 15;          // 0 or 8 (stays in-row for 8 tiles)
    const int nr   = (tile0 >> 4) & 63;   // patch row
    const int ch   = (tile0 >> 10) % 3;
    const int b    = tile0 / 3072;

    float* raw = &smem[wv * 512];         // 4*64 raw coefficients
    float* ym  = raw + 256;               // 16x16 Y scratch (row-major)

    // ---- per-lane constant operands / LDS gather offsets ------------------
    // WMMA A-layout (16x4 f32): lane L holds row m=L%16, VGPR t holds
    //   chunk-local K = t + 2*half.   B-layout (4x16): lane L holds col
    //   n=L%16, VGPR t holds chunk-local row K = t + 2*half.
    v2f dk[4];        // block-diag D16 chunks (A operand of GEMM1 = D16^T
                      // columns; identical values serve as GEMM2 B operand)
    int xo[4][2];     // LDS float offsets for zigzag gather of X (B operand)
#pragma unroll
    for (int kb = 0; kb < 4; ++kb) {
#pragma unroll
        for (int t = 0; t < 2; ++t) {
            const int k = 4 * kb + t + 2 * half;   // global K row/col 0..15
            // D16^T[m=n16][k] == D16[k][n=n16]  (block diagonal, symmetric use)
            const float dv = ((k >> 3) == (n16 >> 3)) ? dctv(k & 7, n16 & 7) : 0.0f;
            if (t == 0) dk[kb].x = dv; else dk[kb].y = dv;
            // X[k][n16] = patch q=(k/8)*2+(n16/8), coeff ZZ[(k%8)][(n16%8)]
            const int q = (k >> 3) * 2 + (n16 >> 3);
            xo[kb][t] = q * 64 + ZZ[(k & 7) * 8 + (n16 & 7)];
        }
    }

    // ---- base pointers (per-tile deltas become instruction immediates) ----
    const int q = lane >> 3, e = lane & 7;            // patch, 8-float slice
    const int n0 = nr * 64 + ncg0 * 4 + q;            // patch index, tile 0
    const float* src = comp + ((size_t)(n0 * 32 + b)) * 192 + ch * 64 + e * 8;
    float* op = out + (((size_t)(b * 3 + ch) * 512 + nr * 8) * 512 + ncg0 * 32)
                    + lane;

    // ---- prefetch tile 0 --------------------------------------------------
    float4 plo[2], phi[2];
    plo[0] = *(const float4*)(src);
    phi[0] = *(const float4*)(src + 4);

#pragma unroll
    for (int i = 0; i < TILES_PER_WAVE; ++i) {
        const int cur = i & 1;
        // issue next tile's global loads before this tile's LDS barrier so
        // HBM latency overlaps the WMMA/LDS pipeline below
        if (i + 1 < TILES_PER_WAVE) {
            plo[cur ^ 1] = *(const float4*)(src + (i + 1) * IN_TILE_STRIDE);
            phi[cur ^ 1] = *(const float4*)(src + (i + 1) * IN_TILE_STRIDE + 4);
        }

        // stage 4 patches (1 KB) into LDS, coalesced b128
        *(float4*)(raw + q * 64 + e * 8)     = plo[cur];
        *(float4*)(raw + q * 64 + e * 8 + 4) = phi[cur];
        // wave-private LDS: DS ops are in-order per wave; wait + fence
        asm volatile("s_wait_dscnt 0x0" ::: "memory");

        // GEMM1: Y = D16^T * X  (zigzag gather into B layout)
        v8f acc = {};
#pragma unroll
        for (int kb = 0; kb < 4; ++kb) {
            v2f x;
            x.x = raw[xo[kb][0]];
            x.y = raw[xo[kb][1]];
            acc = __builtin_amdgcn_wmma_f32_16x16x4_f32(
                false, dk[kb], false, x, (short)0, acc, false, false);
        }

        // spill Y (C layout -> row-major LDS)
#pragma unroll
        for (int v = 0; v < 8; ++v)
            ym[(v + 8 * half) * 16 + n16] = acc[v];
        asm volatile("s_wait_dscnt 0x0" ::: "memory");

        // GEMM2: Z = Y * D16  (Y reloaded in A layout, packed b64)
        v8f z = {};
#pragma unroll
        for (int kb = 0; kb < 4; ++kb) {
            v2f y = *(const v2f*)(ym + n16 * 16 + 4 * kb + 2 * half);
            z = __builtin_amdgcn_wmma_f32_16x16x4_f32(
                false, y, false, dk[kb], (short)0, z, false, false);
        }

        // epilogue: +128, clamp, fully coalesced 128B row stores
        // (lane L -> w offset L: lanes 0-15 = P0|P1 row, 16-31 = P2|P3)
#pragma unroll
        for (int v = 0; v < 8; ++v)
            op[(size_t)v * 512 + i * OUT_TILE_STRIDE] =
                fminf(fmaxf(z[v] + 128.0f, 0.0f), 255.0f);
    }
}

extern "C" void kernel_launch(void* const* d_in, const int* in_sizes, int n_in,
                              void* d_out, int out_size, void* d_ws, size_t ws_size,
                              hipStream_t stream) {
    const float* comp = (const float*)d_in[0];   // (4096, 32, 192) fp32
    float* out = (float*)d_out;                  // (32, 3, 512, 512) fp32
    // 98304 tiles / 8 tiles per wave / 8 waves per 256-thread block
    jpeg_idct_wmma<<<1536, 256, 0, stream>>>(comp, out);
}